// QuantumPolicyHead_11802570129608
// MI455X (gfx1250) — compile-verified
//
#include <hip/hip_runtime.h>
#include <hip/hip_bf16.h>
#include <math.h>

typedef __attribute__((ext_vector_type(2))) float v2f;
typedef __attribute__((ext_vector_type(4))) float v4f;
typedef __attribute__((ext_vector_type(8))) float v8f;
typedef int v4i_vs __attribute__((vector_size(16)));
typedef __attribute__((address_space(1))) v4i_vs* gv4p;   // global int4*
typedef __attribute__((address_space(3))) v4i_vs* lv4p;   // LDS int4*

#define NQ 4
#define HDIM 256
#define NA 16
#define KC 32                // K-chunk staged in LDS per iteration
#define ROWS_PER_WAVE 32
#define WAVES_PER_BLOCK 8
#define ROWS_PER_BLOCK 256
#define TSTRIDE 36           // padded LDS row stride (floats): 16B-aligned, conflict-free

#if __has_builtin(__builtin_amdgcn_global_load_async_to_lds_b128)
#define USE_ASYNC_LDS 1
#endif

__device__ __forceinline__ void wait_async_zero() {
#if defined(USE_ASYNC_LDS)
#if __has_builtin(__builtin_amdgcn_s_wait_asynccnt)
    __builtin_amdgcn_s_wait_asynccnt(0);
#else
    asm volatile("s_wait_asynccnt 0x0" ::: "memory");
#endif
#endif
}

__global__ __launch_bounds__(256)
void qph_fused_kernel(const float* __restrict__ h,
                      const float* __restrict__ Wemb,   // [4][256]
                      const float* __restrict__ bemb,   // [4]
                      const float* __restrict__ qw,     // [2][4][3]
                      const float* __restrict__ Wout,   // [16][4]
                      const float* __restrict__ bout,   // [16]
                      float* __restrict__ out)          // [B][16]
{
    __shared__ float lds_w[NQ * HDIM];                                    // 4 KB
    __shared__ float lds_tile[WAVES_PER_BLOCK][ROWS_PER_WAVE][TSTRIDE];   // 36 KB
    __shared__ float lds_ang[WAVES_PER_BLOCK][ROWS_PER_WAVE][NQ];         // 4 KB

    const int tid  = threadIdx.x;
    const int lane = tid & 31;
    const int wave = tid >> 5;
    const int waveRow = blockIdx.x * ROWS_PER_BLOCK + wave * ROWS_PER_WAVE;

    // Stage W_embed (1024 floats) into LDS: 1 float4 per thread.
    ((v4f*)lds_w)[tid] = ((const v4f*)Wemb)[tid];

    const bool hi   = (lane >= 16);
    const int  nIdx = lane & 15;          // N column (C/D/B) and M-row-in-half (A)
    const int  kOfs = hi ? 2 : 0;         // K sub-offset for A/B fragments
    const bool bAct = (nIdx < NQ);        // lanes carrying real W_embed columns

    v8f c0 = {}; v8f c1 = {};             // accumulators for rows 0-15 / 16-31

    for (int ck = 0; ck < HDIM / KC; ++ck) {
        const int k0 = ck * KC;
        __syncthreads();
        // Stage 32 rows x 32 cols of h -> LDS (128b per lane per iter)
        #pragma unroll
        for (int it = 0; it < 8; ++it) {
            int i  = it * 32 + lane;
            int r  = i >> 3;              // 0..31
            int c4 = i & 7;               // 0..7 (float4 column)
            const float* gp = h + (size_t)(waveRow + r) * HDIM + k0 + c4 * 4;
            float* lp = &lds_tile[wave][r][c4 * 4];
#if defined(USE_ASYNC_LDS)
            __builtin_amdgcn_global_load_async_to_lds_b128((gv4p)gp, (lv4p)lp, 0, 0);
#else
            v4f v = *(const v4f*)gp;
            *(v4f*)lp = v;
#endif
            if (k0 + KC < HDIM) __builtin_prefetch(gp + KC, 0, 0);  // global_prefetch_b8
        }
        wait_async_zero();
        __syncthreads();
        // 8 WMMA k-steps per chunk, two M-tiles each
        #pragma unroll
        for (int kk = 0; kk < KC; kk += 4) {
            v2f a0 = *(const v2f*)&lds_tile[wave][nIdx][kk + kOfs];
            v2f a1 = *(const v2f*)&lds_tile[wave][16 + nIdx][kk + kOfs];
            // B fragment: unconditional safe-row load + cndmask zero (no EXEC churn)
            v2f braw = *(const v2f*)&lds_w[(nIdx & 3) * HDIM + k0 + kk + kOfs];
            v2f b;
            b.x = bAct ? braw.x : 0.f;
            b.y = bAct ? braw.y : 0.f;
            c0 = __builtin_amdgcn_wmma_f32_16x16x4_f32(false, a0, false, b, (short)0, c0, false, false);
            c1 = __builtin_amdgcn_wmma_f32_16x16x4_f32(false, a1, false, b, (short)0, c1, false, false);
        }
    }

    // angles = tanh(pre + b_embed); scatter C-layout -> per-row LDS
    {
        float bq = bemb[nIdx & 3];
        if (bAct) {
            #pragma unroll
            for (int r = 0; r < 8; ++r) {
                int m = r + (hi ? 8 : 0);
                lds_ang[wave][m][nIdx]      = tanhf(c0[r] + bq);
                lds_ang[wave][16 + m][nIdx] = tanhf(c1[r] + bq);
            }
        }
    }
    __syncthreads();

    float theta[NQ];
    #pragma unroll
    for (int q = 0; q < NQ; ++q) theta[q] = lds_ang[wave][lane][q];

    // ---- 4-qubit state-vector sim, fully in registers (lane = batch row) ----
    float sr[16], si[16];
    #pragma unroll
    for (int i = 0; i < 16; ++i) { sr[i] = 0.f; si[i] = 0.f; }
    sr[0] = 1.f;

    // AngleEmbedding: RX(theta[q]) on qubit q  (qubit q <-> bit (3-q))
    #pragma unroll
    for (int q = 0; q < NQ; ++q) {
        float c = cosf(theta[q] * 0.5f);
        float s = sinf(theta[q] * 0.5f);
        const int st = 8 >> q;
        #pragma unroll
        for (int i = 0; i < 16; ++i) {
            if (i & st) continue;
            int j = i + st;
            float a0r = sr[i], a0i = si[i], a1r = sr[j], a1i = si[j];
            sr[i] = c * a0r + s * a1i;  si[i] = c * a0i - s * a1r;
            sr[j] = c * a1r + s * a0i;  si[j] = c * a1i - s * a0r;
        }
    }

    // StronglyEntanglingLayers: Rot + CNOT ring (r = l%3 + 1)
    #pragma unroll
    for (int l = 0; l < 2; ++l) {
        #pragma unroll
        for (int w = 0; w < NQ; ++w) {
            float phi = qw[(l * NQ + w) * 3 + 0];
            float th  = qw[(l * NQ + w) * 3 + 1];
            float om  = qw[(l * NQ + w) * 3 + 2];
            float ct = cosf(th * 0.5f), stn = sinf(th * 0.5f);
            float ap = (phi + om) * 0.5f, am = (phi - om) * 0.5f;
            float cap = cosf(ap), sap = sinf(ap);
            float cam = cosf(am), sam = sinf(am);
            float u00r =  cap * ct,  u00i = -sap * ct;
            float u01r = -cam * stn, u01i = -sam * stn;
            float u10r =  cam * stn, u10i = -sam * stn;
            float u11r =  cap * ct,  u11i =  sap * ct;
            const int st = 8 >> w;
            #pragma unroll
            for (int i = 0; i < 16; ++i) {
                if (i & st) continue;
                int j = i + st;
                float a0r = sr[i], a0i = si[i], a1r = sr[j], a1i = si[j];
                sr[i] = u00r*a0r - u00i*a0i + u01r*a1r - u01i*a1i;
                si[i] = u00r*a0i + u00i*a0r + u01r*a1i + u01i*a1r;
                sr[j] = u10r*a0r - u10i*a0i + u11r*a1r - u11i*a1i;
                si[j] = u10r*a0i + u10i*a0r + u11r*a1i + u11i*a1r;
            }
        }
        const int rr = (l % 3) + 1;
        #pragma unroll
        for (int w = 0; w < NQ; ++w) {
            const int t   = (w + rr) & 3;
            const int stc = 8 >> w, stt = 8 >> t;
            #pragma unroll
            for (int i = 0; i < 16; ++i) {
                if ((i & stc) && !(i & stt)) {
                    int j = i | stt;
                    float tr = sr[i], ti = si[i];
                    sr[i] = sr[j]; si[i] = si[j];
                    sr[j] = tr;    si[j] = ti;
                }
            }
        }
    }

    // PauliZ expvals
    float z[NQ] = {0.f, 0.f, 0.f, 0.f};
    #pragma unroll
    for (int i = 0; i < 16; ++i) {
        float p = sr[i] * sr[i] + si[i] * si[i];
        #pragma unroll
        for (int q = 0; q < NQ; ++q)
            z[q] += ((i >> (3 - q)) & 1) ? -p : p;
    }

    // ---- Output head: out = z @ W_out^T + b_out, one WMMA per 16-row tile ----
    v2f bo;                                   // B[k][n] = W_out[n][k]
    bo.x = Wout[nIdx * NQ + kOfs];
    bo.y = Wout[nIdx * NQ + kOfs + 1];
    float bias = bout[nIdx];
    v8f d0, d1;
    #pragma unroll
    for (int r = 0; r < 8; ++r) { d0[r] = bias; d1[r] = bias; }

    float z0x = __shfl_xor(z[0], 16, 32);
    float z1x = __shfl_xor(z[1], 16, 32);
    float z2x = __shfl_xor(z[2], 16, 32);
    float z3x = __shfl_xor(z[3], 16, 32);
    v2f at0, at1;
    if (!hi) { at0.x = z[0]; at0.y = z[1]; at1.x = z0x;  at1.y = z1x;  }
    else     { at0.x = z2x;  at0.y = z3x;  at1.x = z[2]; at1.y = z[3]; }

    d0 = __builtin_amdgcn_wmma_f32_16x16x4_f32(false, at0, false, bo, (short)0, d0, false, false);
    d1 = __builtin_amdgcn_wmma_f32_16x16x4_f32(false, at1, false, bo, (short)0, d1, false, false);

    #pragma unroll
    for (int r = 0; r < 8; ++r) {
        int m = r + (hi ? 8 : 0);
        out[(size_t)(waveRow + m) * NA + nIdx]      = d0[r];
        out[(size_t)(waveRow + 16 + m) * NA + nIdx] = d1[r];
    }
}

extern "C" void kernel_launch(void* const* d_in, const int* in_sizes, int n_in,
                              void* d_out, int out_size, void* d_ws, size_t ws_size,
                              hipStream_t stream) {
    const float* h    = (const float*)d_in[0];
    const float* Wemb = (const float*)d_in[1];
    const float* bemb = (const float*)d_in[2];
    const float* qw   = (const float*)d_in[3];
    const float* Wout = (const float*)d_in[4];
    const float* bout = (const float*)d_in[5];
    float* out = (float*)d_out;

    const int Btot = in_sizes[0] / HDIM;          // 262144
    dim3 grid(Btot / ROWS_PER_BLOCK), block(256); // exact cover, EXEC all-ones
    qph_fused_kernel<<<grid, block, 0, stream>>>(h, Wemb, bemb, qw, Wout, bout, out);
}